// HuberRankLoss_19121194402346
// MI455X (gfx1250) — compile-verified
//
#include <hip/hip_runtime.h>

// HuberRankLoss on gfx1250 (MI455X): B=32, N=1024.
// Pairwise rank loss as 16x16 tiles; outer differences generated by
// V_WMMA_F32_16X16X4_F32 (rank-2 matmul trick). Nonlinearity in VALU with
// per-element accumulators (no packing movs); pair counting moved to SALU
// via popc(ballot()) so it co-issues with the VALU/WMMA stream.

typedef __attribute__((ext_vector_type(2))) float v2f;
typedef __attribute__((ext_vector_type(8))) float v8f;

#define B_SZ    32
#define N_SZ    1024
#define JGROUPS 8            // blocks per batch (8 waves * 16 cols = 128 cols each)
#define TILE    16
#define ITILES  (N_SZ / TILE) // 64

__global__ __launch_bounds__(256)
void hr_pair_kernel(const float* __restrict__ pred,
                    const float* __restrict__ target,
                    const int*   __restrict__ mask,
                    float*       __restrict__ ws)
{
    __shared__ float s_trow[N_SZ]; // target, masked rows -> +1e30
    __shared__ float s_tcol[N_SZ]; // target, masked cols -> -1e30
    __shared__ float s_p[N_SZ];    // pred (raw)

    const int tid    = threadIdx.x;
    const int lane   = tid & 31;
    const int wave   = tid >> 5;
    const int batch  = blockIdx.x >> 3;
    const int jgroup = blockIdx.x & 7;

    // ---- stage batch into LDS; fold Huber into staging (jgroup==0 only) ----
    float hub = 0.0f, mcnt = 0.0f;
    for (int idx = tid; idx < N_SZ; idx += 256) {
        const int   g = batch * N_SZ + idx;
        const float p = pred[g];
        const float t = target[g];
        const float m = (mask[g] != 0) ? 1.0f : 0.0f;
        s_p[idx]    = p;
        s_trow[idx] = (m != 0.0f) ? t :  1e30f;
        s_tcol[idx] = (m != 0.0f) ? t : -1e30f;
        if (jgroup == 0) {
            const float e  = p - t;
            const float ae = fabsf(e);
            const float h  = (ae <= 1.0f) ? (0.5f * e * e) : (ae - 0.5f);
            hub  += h * m;
            mcnt += m;
        }
    }
    __syncthreads();

    // ---- per-wave fixed B operands (16 columns j) ----
    // B layout (4x16, 2 VGPRs): vgpr0 = {K0 lanes0-15, K2 lanes16-31},
    //                           vgpr1 = {K1 lanes0-15, K3 lanes16-31}.
    const int   jcol = jgroup * 128 + wave * TILE + (lane & 15);
    const float low  = (lane < 16) ? 1.0f : 0.0f; // zeroes K2/K3 pad rows
    const float nlow = -low;                      // hoisted loop-invariant
    const float tc   = s_tcol[jcol];
    const float pc   = s_p[jcol];
    v2f b1; b1.x = tc * low; b1.y = low;          // dt: [t_col ; 1 ; 0 ; 0]
    v2f b2; b2.x = low;      b2.y = pc * low;     // dp: [1 ; p_col ; 0 ; 0]
    const v8f cz = {0.f, 0.f, 0.f, 0.f, 0.f, 0.f, 0.f, 0.f};

    // 8 independent loss accumulators (no cross-element dependency chain).
    float lacc[8];
#pragma unroll
    for (int e = 0; e < 8; ++e) lacc[e] = 0.0f;
    unsigned ucnt = 0; // wave-uniform pair count (SALU: s_bcnt + s_add)

    // ---- sweep 64 row tiles: 2 WMMA + elementwise select per 256 pairs ----
    for (int it = 0; it < ITILES; ++it) {
        const int   irow = it * TILE + (lane & 15);
        const float tr   = s_trow[irow];
        const float pr   = s_p[irow];
        // A layout (16x4, 2 VGPRs): lanes0-15 = {K0,K1}, lanes16-31 = {K2,K3}=0.
        v2f a1; a1.x = low;      a1.y = -tr * low; // dt[i][j] = t_j - t_i
        v2f a2; a2.x = pr * low; a2.y = nlow;      // dp[i][j] = p_i - p_j
        v8f dt = __builtin_amdgcn_wmma_f32_16x16x4_f32(
                     false, a1, false, b1, (short)0, cz, false, false);
        v8f dp = __builtin_amdgcn_wmma_f32_16x16x4_f32(
                     false, a2, false, b2, (short)0, cz, false, false);
#pragma unroll
        for (int e = 0; e < 8; ++e) {
            const bool valid = dt[e] > 0.0f;                  // == pair_valid
            ucnt += __popc((unsigned)__ballot(valid));        // SALU count
            lacc[e] += valid ? fmaxf(dp[e], 0.0f) : 0.0f;     // relu(p_i - p_j)
        }
    }

    // ---- reductions ----
    float lsum = ((lacc[0] + lacc[1]) + (lacc[2] + lacc[3]))
               + ((lacc[4] + lacc[5]) + (lacc[6] + lacc[7]));
#pragma unroll
    for (int s = 16; s >= 1; s >>= 1) {
        lsum += __shfl_xor(lsum, s, 32);
        hub  += __shfl_xor(hub,  s, 32);
        mcnt += __shfl_xor(mcnt, s, 32);
    }
    if (lane == 0) {
        atomicAdd(&ws[2], lsum);
        atomicAdd(&ws[3], (float)ucnt); // uniform across the wave
        if (jgroup == 0) {
            atomicAdd(&ws[0], hub);
            atomicAdd(&ws[1], mcnt);
        }
    }
}

__global__ void hr_final_kernel(const float* __restrict__ ws,
                                float* __restrict__ out)
{
    if (threadIdx.x == 0 && blockIdx.x == 0) {
        const float huber = ws[0] / fmaxf(ws[1], 1.0f);
        const float rank  = ws[2] / fmaxf(ws[3], 1.0f);
        out[0] = huber + 0.5f * rank;
    }
}

extern "C" void kernel_launch(void* const* d_in, const int* in_sizes, int n_in,
                              void* d_out, int out_size, void* d_ws, size_t ws_size,
                              hipStream_t stream)
{
    const float* pred   = (const float*)d_in[0];
    const float* target = (const float*)d_in[1];
    const int*   mask   = (const int*)d_in[2];
    float*       ws     = (float*)d_ws;
    float*       out    = (float*)d_out;

    hipMemsetAsync(ws, 0, 4 * sizeof(float), stream);
    hr_pair_kernel<<<B_SZ * JGROUPS, 256, 0, stream>>>(pred, target, mask, ws);
    hr_final_kernel<<<1, 32, 0, stream>>>(ws, out);
}